// UGCN_20469814133410
// MI455X (gfx1250) — compile-verified
//
#include <hip/hip_runtime.h>
#include <math.h>

#define N_NODES 20000
#define F_IN    500
#define HID     64
#define CLS     16
#define NHEAD   4
#define SLOPE   0.2f

typedef __attribute__((ext_vector_type(2))) float v2f;
typedef __attribute__((ext_vector_type(8))) float v8f;
typedef int i32x4 __attribute__((vector_size(16)));   // matches builtin param

#if defined(__has_builtin)
#if __has_builtin(__builtin_amdgcn_global_load_async_to_lds_b128) && \
    __has_builtin(__builtin_amdgcn_s_wait_asynccnt)
#define USE_ASYNC_LDS 1
#endif
#endif

// 16-byte global -> LDS copy. Async (no VGPR round trip, ASYNCcnt-tracked)
// when the toolchain exposes the gfx1250 builtin, else float4 via VGPRs.
__device__ __forceinline__ void cp16(const float* __restrict__ g, float* l) {
#ifdef USE_ASYNC_LDS
    __builtin_amdgcn_global_load_async_to_lds_b128(
        (__attribute__((address_space(1))) i32x4*)(g),
        (__attribute__((address_space(3))) i32x4*)(l), 0, 0);
#else
    *(float4*)l = *(const float4*)g;
#endif
}

// ---------------------------------------------------------------------------
// fp32 WMMA GEMM:  Z[N,C] = X[N,K] @ W[K,C]
// block = 128 threads (4 waves); block computes a 16(M) x 64(N) tile; wave w
// owns columns [w*16, w*16+16). K staged in chunks of 64 through a double-
// buffered LDS pipeline fed by GLOBAL_LOAD_ASYNC_TO_LDS_B128.
// grid = (N/16, C/64). Requires N%16==0, C%64==0, K>=4.
// ---------------------------------------------------------------------------
#define TK 64
#define LDA 68   // floats; 272B rows: 16B-aligned + conflict-free column reads
__global__ __launch_bounds__(128)
void gemm_wmma_f32(const float* __restrict__ X, const float* __restrict__ W,
                   float* __restrict__ Z, int N, int K, int C) {
    __shared__ float As[2][16][LDA];
    __shared__ float Bs[2][TK][LDA];

    const int tid  = threadIdx.x;
    const int wave = tid >> 5;
    const int lane = tid & 31;
    const int rowBase = blockIdx.x * 16;
    const int colBase = blockIdx.y * 64;

    const int lrow  = lane & 15;          // M (A frag) / N (B frag) in tile
    const int khalf = (lane >> 4) << 1;   // 0 for lanes 0-15, 2 for 16-31
    const int wcol  = wave * 16;

    // issue one tile stage: exactly 10 async b128 ops per wave (2 A + 8 B),
    // uniform control flow; K-tail handled by in-range address clamping.
    auto stage = [&](int b, int k0s) {
        #pragma unroll
        for (int j = 0; j < 2; ++j) {               // A: 16x64 = 256 chunks
            int cl = tid + 128 * j;
            int r = cl >> 4, c4 = (cl & 15) << 2;
            int gc = min(k0s + c4, K - 4);
            cp16(X + (size_t)(rowBase + r) * K + gc, &As[b][r][c4]);
        }
        #pragma unroll
        for (int j = 0; j < 8; ++j) {               // B: 64x64 = 1024 chunks
            int cl = tid + 128 * j;
            int r = cl >> 4, c4 = (cl & 15) << 2;
            int gk = min(k0s + r, K - 1);
            cp16(W + (size_t)gk * C + colBase + c4, &Bs[b][r][c4]);
        }
    };

    v8f acc = {};
    stage(0, 0);
    int buf = 0;

    for (int k0 = 0; k0 < K; k0 += TK) {
        const bool have_next = (k0 + TK) < K;
        if (have_next) stage(buf ^ 1, k0 + TK);

#ifdef USE_ASYNC_LDS
        // async loads complete in order: <=10 outstanding means the current
        // buffer's 10 ops have landed while the next tile stays in flight.
        if (have_next) __builtin_amdgcn_s_wait_asynccnt(10);
        else           __builtin_amdgcn_s_wait_asynccnt(0);
#endif
        const int klim = K - k0;                    // < TK only on last tile
        if (klim < TK) {                            // zero clamped K-tail
            for (int i = tid; i < 16 * TK; i += 128) {
                int r = i >> 6, c = i & 63;
                if (c >= klim) As[buf][r][c] = 0.0f;
            }
            for (int i = tid; i < TK * 64; i += 128) {
                int r = i >> 6, c = i & 63;
                if (r >= klim) Bs[buf][r][c] = 0.0f;
            }
        }
        __syncthreads();

        #pragma unroll
        for (int kk = 0; kk < TK; kk += 4) {
            v2f a, b;
            a.x = As[buf][lrow][kk + khalf + 0];
            a.y = As[buf][lrow][kk + khalf + 1];
            b.x = Bs[buf][kk + khalf + 0][wcol + lrow];
            b.y = Bs[buf][kk + khalf + 1][wcol + lrow];
            acc = __builtin_amdgcn_wmma_f32_16x16x4_f32(
                false, a, false, b, (short)0, acc, false, false);
        }
        __syncthreads();                            // buf free for re-stage
        buf ^= 1;
    }

    // C/D layout: VGPR i -> row i (lanes 0-15) / row i+8 (lanes 16-31)
    const int rplus = (lane >> 4) << 3;
    const int col   = colBase + wcol + lrow;
    #pragma unroll
    for (int i = 0; i < 8; i++) {
        Z[(size_t)(rowBase + i + rplus) * C + col] = acc[i];
    }
}

// ---------------------------------------------------------------------------
// helpers
// ---------------------------------------------------------------------------
__global__ void fill_f32(float* p, float v, int n) {
    int i = blockIdx.x * blockDim.x + threadIdx.x;
    if (i < n) p[i] = v;
}

__device__ __forceinline__ void atomicMaxF32(float* addr, float v) {
    if (v >= 0.0f) atomicMax((int*)addr, __float_as_int(v));
    else           atomicMin((unsigned int*)addr, __float_as_uint(v));
}

// e_src[n,h] = sum_d z[n,h,d]*a_s[h,d] ; e_dst analogous
__global__ void attn_scores(const float* __restrict__ Z,
                            const float* __restrict__ a_s,
                            const float* __restrict__ a_d,
                            float* __restrict__ es, float* __restrict__ ed,
                            int n_total, int D) {
    int idx = blockIdx.x * blockDim.x + threadIdx.x;   // n*H + h
    if (idx >= n_total) return;
    int h = idx & (NHEAD - 1);
    const float* zr = Z + (size_t)idx * D;
    const float* as = a_s + h * D;
    const float* ad = a_d + h * D;
    float s1 = 0.f, s2 = 0.f;
    for (int d = 0; d < D; d++) { float z = zr[d]; s1 += z * as[d]; s2 += z * ad[d]; }
    es[idx] = s1; ed[idx] = s2;
}

__device__ __forceinline__ float edge_e(const int* ei, int E,
                                        const float* es, const float* ed,
                                        int e, int h) {
    int s = ei[e], d = ei[E + e];
    float v = es[s * NHEAD + h] + ed[d * NHEAD + h];
    return (v > 0.f) ? v : SLOPE * v;
}

__global__ void edge_segmax(const int* __restrict__ ei, int E,
                            const float* __restrict__ es, const float* __restrict__ ed,
                            float* __restrict__ m) {
    int idx = blockIdx.x * blockDim.x + threadIdx.x;
    if (idx >= E * NHEAD) return;
    int e = idx >> 2, h = idx & (NHEAD - 1);
    float v = edge_e(ei, E, es, ed, e, h);
    atomicMaxF32(&m[ei[E + e] * NHEAD + h], v);
}

__global__ void edge_expsum(const int* __restrict__ ei, int E,
                            const float* __restrict__ es, const float* __restrict__ ed,
                            const float* __restrict__ m,
                            float* __restrict__ pbuf, float* __restrict__ ssum) {
    int idx = blockIdx.x * blockDim.x + threadIdx.x;
    if (idx >= E * NHEAD) return;
    int e = idx >> 2, h = idx & (NHEAD - 1);
    int d = ei[E + e];
    float v = edge_e(ei, E, es, ed, e, h);
    float p = expf(v - m[d * NHEAD + h]);
    pbuf[idx] = p;
    atomicAdd(&ssum[d * NHEAD + h], p);
}

// out[dst,h,:] += alpha * z[src,h,:]   (one thread per (edge,head))
__global__ void edge_aggregate(const int* __restrict__ ei, int E,
                               const float* __restrict__ pbuf,
                               const float* __restrict__ ssum,
                               const float* __restrict__ Z,
                               float* __restrict__ agg, int D) {
    int idx = blockIdx.x * blockDim.x + threadIdx.x;
    if (idx >= E * NHEAD) return;
    int e = idx >> 2, h = idx & (NHEAD - 1);
    int s = ei[e], d = ei[E + e];
    float alpha = pbuf[idx] / (ssum[d * NHEAD + h] + 1e-16f);
    const float* zr = Z   + ((size_t)s * NHEAD + h) * D;
    float*       ar = agg + ((size_t)d * NHEAD + h) * D;
    for (int k = 0; k < D; k++) atomicAdd(&ar[k], alpha * zr[k]);
}

// out[n,d] = mean_h(agg[n,h,d]) + b[d] ; optional relu
__global__ void head_mean_bias(const float* __restrict__ agg,
                               const float* __restrict__ b,
                               float* __restrict__ out, int D, int relu) {
    int idx = blockIdx.x * blockDim.x + threadIdx.x;   // n*D + d
    if (idx >= N_NODES * D) return;
    int n = idx / D, d = idx - n * D;
    const float* ar = agg + (size_t)n * NHEAD * D + d;
    float s = 0.f;
    #pragma unroll
    for (int h = 0; h < NHEAD; h++) s += ar[h * D];
    s = s * (1.0f / NHEAD) + b[d];
    if (relu) s = fmaxf(s, 0.f);
    out[idx] = s;
}

// final 3-branch attention fusion + log_softmax. 16 lanes per node.
__global__ void fuse_branches(const float* __restrict__ h1, const float* __restrict__ h2,
                              const float* __restrict__ h3,
                              const float* __restrict__ l1w, const float* __restrict__ l1b,
                              const float* __restrict__ l2w, const float* __restrict__ l2b,
                              const float* __restrict__ l3w, const float* __restrict__ l3b,
                              const float* __restrict__ aggw,
                              float* __restrict__ out) {
    int tid = blockIdx.x * blockDim.x + threadIdx.x;
    int n = tid >> 4, c = tid & 15;
    if (n >= N_NODES) return;

    const float* hs[3] = {h1, h2, h3};
    const float* lw[3] = {l1w, l2w, l3w};
    const float* lb[3] = {l1b, l2b, l3b};

    float a[3];
    #pragma unroll
    for (int i = 0; i < 3; i++) {
        const float* hr = hs[i] + (size_t)n * CLS;
        float t = lb[i][c];
        #pragma unroll
        for (int k = 0; k < CLS; k++) t += hr[k] * lw[i][k * CLS + c];
        t = tanhf(t) * aggw[c];
        #pragma unroll
        for (int off = 8; off >= 1; off >>= 1) t += __shfl_xor(t, off, 16);
        a[i] = t;
    }
    float mx = fmaxf(a[0], fmaxf(a[1], a[2]));
    float e0 = expf(a[0] - mx), e1 = expf(a[1] - mx), e2 = expf(a[2] - mx);
    float inv = 1.0f / (e0 + e1 + e2);
    size_t off16 = (size_t)n * CLS + c;
    float hv = (e0 * h1[off16] + e1 * h2[off16] + e2 * h3[off16]) * inv;

    float m = hv;
    #pragma unroll
    for (int off = 8; off >= 1; off >>= 1) m = fmaxf(m, __shfl_xor(m, off, 16));
    float ex = expf(hv - m), ss = ex;
    #pragma unroll
    for (int off = 8; off >= 1; off >>= 1) ss += __shfl_xor(ss, off, 16);
    out[off16] = hv - m - logf(ss);
}

// ---------------------------------------------------------------------------
// orchestration
// ---------------------------------------------------------------------------
static inline int cdiv(int a, int b) { return (a + b - 1) / b; }

extern "C" void kernel_launch(void* const* d_in, const int* in_sizes, int n_in,
                              void* d_out, int out_size, void* d_ws, size_t ws_size,
                              hipStream_t stream) {
    const float* x = (const float*)d_in[0];
    const int* ei[3]  = {(const int*)d_in[1], (const int*)d_in[2], (const int*)d_in[3]};
    const int  Ecnt[3] = {in_sizes[1] / 2, in_sizes[2] / 2, in_sizes[3] / 2};

    // workspace layout (floats)
    float* ws   = (float*)d_ws;
    float* zA   = ws;                       // N*256
    float* aggA = zA   + (size_t)N_NODES * 256;
    float* pE   = aggA + (size_t)N_NODES * 256;          // maxE*4
    float* esb  = pE   + (size_t)660000 * NHEAD;
    float* edb  = esb  + (size_t)N_NODES * NHEAD;
    float* mb   = edb  + (size_t)N_NODES * NHEAD;
    float* sb   = mb   + (size_t)N_NODES * NHEAD;
    float* hA   = sb   + (size_t)N_NODES * NHEAD;        // N*64
    float* zB   = hA   + (size_t)N_NODES * HID;          // N*64
    float* aggB = zB   + (size_t)N_NODES * HID;          // N*64
    float* hbr  = aggB + (size_t)N_NODES * HID;          // 3 * N*16

    const int BLK = 256;

    for (int br = 0; br < 3; br++) {
        const int E = Ecnt[br];
        const float* WA = (const float*)d_in[4 + br * 8 + 0];
        const float* sA = (const float*)d_in[4 + br * 8 + 1];
        const float* dA = (const float*)d_in[4 + br * 8 + 2];
        const float* bA = (const float*)d_in[4 + br * 8 + 3];
        const float* WB = (const float*)d_in[4 + br * 8 + 4];
        const float* sB = (const float*)d_in[4 + br * 8 + 5];
        const float* dB = (const float*)d_in[4 + br * 8 + 6];
        const float* bB = (const float*)d_in[4 + br * 8 + 7];

        // ---- layer A: z = x @ WA  [N, 4*64] ----
        gemm_wmma_f32<<<dim3(N_NODES / 16, (NHEAD * HID) / 64), 128, 0, stream>>>(
            x, WA, zA, N_NODES, F_IN, NHEAD * HID);
        attn_scores<<<cdiv(N_NODES * NHEAD, BLK), BLK, 0, stream>>>(
            zA, sA, dA, esb, edb, N_NODES * NHEAD, HID);
        fill_f32<<<cdiv(N_NODES * NHEAD, BLK), BLK, 0, stream>>>(mb, -INFINITY, N_NODES * NHEAD);
        fill_f32<<<cdiv(N_NODES * NHEAD, BLK), BLK, 0, stream>>>(sb, 0.0f, N_NODES * NHEAD);
        edge_segmax<<<cdiv(E * NHEAD, BLK), BLK, 0, stream>>>(ei[br], E, esb, edb, mb);
        edge_expsum<<<cdiv(E * NHEAD, BLK), BLK, 0, stream>>>(ei[br], E, esb, edb, mb, pE, sb);
        fill_f32<<<cdiv(N_NODES * NHEAD * HID, BLK), BLK, 0, stream>>>(aggA, 0.0f, N_NODES * NHEAD * HID);
        edge_aggregate<<<cdiv(E * NHEAD, BLK), BLK, 0, stream>>>(ei[br], E, pE, sb, zA, aggA, HID);
        head_mean_bias<<<cdiv(N_NODES * HID, BLK), BLK, 0, stream>>>(aggA, bA, hA, HID, 1);

        // ---- layer B: z = hA @ WB  [N, 4*16] ----
        gemm_wmma_f32<<<dim3(N_NODES / 16, (NHEAD * CLS) / 64), 128, 0, stream>>>(
            hA, WB, zB, N_NODES, HID, NHEAD * CLS);
        attn_scores<<<cdiv(N_NODES * NHEAD, BLK), BLK, 0, stream>>>(
            zB, sB, dB, esb, edb, N_NODES * NHEAD, CLS);
        fill_f32<<<cdiv(N_NODES * NHEAD, BLK), BLK, 0, stream>>>(mb, -INFINITY, N_NODES * NHEAD);
        fill_f32<<<cdiv(N_NODES * NHEAD, BLK), BLK, 0, stream>>>(sb, 0.0f, N_NODES * NHEAD);
        edge_segmax<<<cdiv(E * NHEAD, BLK), BLK, 0, stream>>>(ei[br], E, esb, edb, mb);
        edge_expsum<<<cdiv(E * NHEAD, BLK), BLK, 0, stream>>>(ei[br], E, esb, edb, mb, pE, sb);
        fill_f32<<<cdiv(N_NODES * NHEAD * CLS, BLK), BLK, 0, stream>>>(aggB, 0.0f, N_NODES * NHEAD * CLS);
        edge_aggregate<<<cdiv(E * NHEAD, BLK), BLK, 0, stream>>>(ei[br], E, pE, sb, zB, aggB, CLS);
        head_mean_bias<<<cdiv(N_NODES * CLS, BLK), BLK, 0, stream>>>(
            aggB, bB, hbr + (size_t)br * N_NODES * CLS, CLS, 0);
    }

    fuse_branches<<<cdiv(N_NODES * 16, BLK), BLK, 0, stream>>>(
        hbr, hbr + (size_t)N_NODES * CLS, hbr + (size_t)2 * N_NODES * CLS,
        (const float*)d_in[28], (const float*)d_in[29],
        (const float*)d_in[30], (const float*)d_in[31],
        (const float*)d_in[32], (const float*)d_in[33],
        (const float*)d_in[34], (float*)d_out);
}